// AMNet_44796508897550
// MI455X (gfx1250) — compile-verified
//
#include <hip/hip_runtime.h>
#include <math.h>

#define IN_F 166
#define HID  156
#define FNUM 4
#define NPAD 160            // HID padded to multiple of 16
#define NT   10             // N-tiles (16 cols each) per wave

typedef float v2f __attribute__((ext_vector_type(2)));
typedef float v8f __attribute__((ext_vector_type(8)));

// ---------------------------------------------------------------- utilities
__global__ void fill_zero_kernel(float* __restrict__ p, size_t n) {
  size_t i = (size_t)blockIdx.x * blockDim.x + threadIdx.x;
  size_t stride = (size_t)gridDim.x * blockDim.x;
  for (; i < n; i += stride) p[i] = 0.0f;
}

__global__ void degree_kernel(const long long* __restrict__ ei, float* __restrict__ deg, int E) {
  int i = blockIdx.x * blockDim.x + threadIdx.x;
  int stride = gridDim.x * blockDim.x;
  for (; i < E; i += stride) {
    int dst = (int)ei[(size_t)E + i];
    atomicAdd(&deg[dst], 1.0f);
  }
}

__global__ void dinv_kernel(float* __restrict__ deg, int n) {
  int i = blockIdx.x * blockDim.x + threadIdx.x;
  int stride = gridDim.x * blockDim.x;
  for (; i < n; i += stride) deg[i] = rsqrtf(fmaxf(deg[i], 1.0f));
}

// ---------------------------------------------------------------- weight packing
// B [K,Nn] row-major  ->  Bp [pairs][NPAD][2] : K-pair interleaved, zero padded.
// Lane layout for V_WMMA_F32_16X16X4_F32 B-operand: one b64 load per lane.
__global__ void pack_b_kernel(const float* __restrict__ B, float* __restrict__ Bp,
                              int K, int Nn, int pairs) {
  int i = blockIdx.x * blockDim.x + threadIdx.x;
  int stride = gridDim.x * blockDim.x;
  int total = pairs * NPAD;
  for (; i < total; i += stride) {
    int kp = i / NPAD, n = i % NPAD;
    int k0 = 2 * kp;
    float v0 = (k0     < K && n < Nn) ? B[(size_t)k0       * Nn + n] : 0.0f;
    float v1 = (k0 + 1 < K && n < Nn) ? B[(size_t)(k0 + 1) * Nn + n] : 0.0f;
    Bp[(size_t)i * 2 + 0] = v0;
    Bp[(size_t)i * 2 + 1] = v1;
  }
}

// ------------------------------------------------- sparse propagation (A_hat @ h)
__global__ void prop_kernel(const float* __restrict__ hin, float* __restrict__ out,
                            const long long* __restrict__ ei, const float* __restrict__ dinv,
                            int E) {
  int lane = threadIdx.x & 31;
  int wid  = (blockIdx.x * blockDim.x + threadIdx.x) >> 5;
  int nw   = (gridDim.x * blockDim.x) >> 5;
  for (int e = wid; e < E; e += nw) {
    int src = (int)ei[e];
    int dst = (int)ei[(size_t)E + e];
    float w = dinv[src] * dinv[dst];
    const float* hs = hin + (size_t)src * HID;
    float* od = out + (size_t)dst * HID;
    for (int d = lane; d < HID; d += 32)
      atomicAdd(&od[d], w * hs[d]);
  }
}

// ------------------------------------------------- fp32 WMMA GEMM
// C[M,HID] = A[M,K] @ B (packed Bp) (+bias)(+relu).
// One wave owns a 16-row x 160-col strip (10 accumulators); 8 waves/block = 128 rows.
// Steady-state inner loop: 1 A b64-load + 10 B b64-loads + 10 v_wmma, no guards.
__global__ __launch_bounds__(256) void gemm_wmma(
    const float* __restrict__ A, int lda, int K,
    const float* __restrict__ Bp, const float* __restrict__ bias,
    float* __restrict__ C, int M, int relu) {
  const int lane = threadIdx.x & 31;
  const int wave = threadIdx.x >> 5;
  const int half = lane >> 4;           // selects K pair {0,1} vs {2,3}
  const int l16  = lane & 15;
  const int m0 = (blockIdx.x * 8 + wave) * 16;
  if (m0 >= M) return;                  // uniform per wave; EXEC all-ones below
  int mA = m0 + l16;
  if (mA > M - 1) mA = M - 1;           // row clamp instead of masked loads
  const float* Arow = A + (size_t)mA * lda;

  v8f acc[NT] = {};

  const int kmain = K & ~3;
  int k0 = 0;
  for (; k0 < kmain; k0 += 4) {
    v2f a = *(const v2f*)(Arow + k0 + 2 * half);
    const float* bp = Bp + ((size_t)(k0 >> 1) + half) * (NPAD * 2);
#pragma unroll
    for (int t = 0; t < NT; ++t) {
      v2f b = *(const v2f*)(bp + (t * 16 + l16) * 2);
      acc[t] = __builtin_amdgcn_wmma_f32_16x16x4_f32(false, a, false, b,
                                                     (short)0, acc[t], false, false);
    }
  }
  if (k0 < K) {                         // K tail (K=166 only): select, Bp rows are zero-padded
    const int ka = k0 + 2 * half;
    v2f a;
    a.x = (ka     < K) ? Arow[ka]     : 0.0f;
    a.y = (ka + 1 < K) ? Arow[ka + 1] : 0.0f;
    const float* bp = Bp + ((size_t)(k0 >> 1) + half) * (NPAD * 2);
#pragma unroll
    for (int t = 0; t < NT; ++t) {
      v2f b = *(const v2f*)(bp + (t * 16 + l16) * 2);
      acc[t] = __builtin_amdgcn_wmma_f32_16x16x4_f32(false, a, false, b,
                                                     (short)0, acc[t], false, false);
    }
  }

#pragma unroll
  for (int t = 0; t < NT; ++t) {
    const int n = t * 16 + l16;
    if (n < HID) {
      const float bv = bias ? bias[n] : 0.0f;
#pragma unroll
      for (int i = 0; i < 8; ++i) {
        int m = m0 + i + 8 * half;      // C/D layout: VGPR i -> row i (+8 upper half)
        if (m < M) {
          float v = acc[t][i] + bv;
          if (relu) v = fmaxf(v, 0.0f);
          C[(size_t)m * HID + n] = v;
        }
      }
    }
  }
}

// ------------------------------------------------- filter coefficients + rowf
__global__ void coef_kernel(const float* __restrict__ theta, const float* __restrict__ b_filt,
                            const float* __restrict__ Wf, const float* __restrict__ bfv,
                            float* __restrict__ abc, float* __restrict__ rowf) {
  int tid = threadIdx.x;
  if (tid < FNUM) {
    float c0 = fmaxf(theta[tid * 3 + 0], 0.0f);
    float c1 = fmaxf(theta[tid * 3 + 1], 0.0f);
    float c2 = fmaxf(theta[tid * 3 + 2], 0.0f);
    abc[tid * 3 + 0] = 0.25f * c0 + 0.5f * c1 + 0.25f * c2;  // alpha (h)
    abc[tid * 3 + 1] = 0.5f  * c0 - 0.5f * c2;               // beta  (Ph)
    abc[tid * 3 + 2] = 0.25f * c0 - 0.5f * c1 + 0.25f * c2;  // gamma (P^2 h)
  }
  for (int i = tid; i < FNUM * HID; i += blockDim.x) {
    int f = i / HID, d = i % HID;
    float s = bfv[d];
    for (int j = 0; j < HID; ++j) s += b_filt[f * HID + j] * Wf[j * HID + d];
    rowf[i] = s;
  }
}

// ------------------------------------------------- attention fusion + head (wave/node)
__global__ void final_kernel(const float* __restrict__ Gh, const float* __restrict__ G1,
                             const float* __restrict__ G2, const float* __restrict__ Xp,
                             const float* __restrict__ h,  const float* __restrict__ p1,
                             const float* __restrict__ p2, const float* __restrict__ abc,
                             const float* __restrict__ rowf, const float* __restrict__ b_filt,
                             const float* __restrict__ Wc, const float* __restrict__ bc,
                             float* __restrict__ out, int N) {
  int lane = threadIdx.x & 31;
  int wid  = (blockIdx.x * blockDim.x + threadIdx.x) >> 5;
  int nw   = (gridDim.x * blockDim.x) >> 5;

  float af[FNUM], be[FNUM], ga[FNUM];
#pragma unroll
  for (int f = 0; f < FNUM; ++f) {
    af[f] = abc[f * 3 + 0]; be[f] = abc[f * 3 + 1]; ga[f] = abc[f * 3 + 2];
  }

  for (int n = wid; n < N; n += nw) {
    size_t base = (size_t)n * HID;
    float lg[FNUM] = {0.0f, 0.0f, 0.0f, 0.0f};
    for (int d = lane; d < HID; d += 32) {
      float gh = Gh[base + d], g1 = G1[base + d], g2 = G2[base + d];
      float xp = tanhf(Xp[base + d]);
#pragma unroll
      for (int f = 0; f < FNUM; ++f) {
        float hp = tanhf(af[f] * gh + be[f] * g1 + ga[f] * g2 + rowf[f * HID + d]);
        lg[f] += hp * xp;
      }
    }
#pragma unroll
    for (int f = 0; f < FNUM; ++f)
      for (int off = 16; off > 0; off >>= 1)
        lg[f] += __shfl_xor(lg[f], off, 32);

    float mx = fmaxf(fmaxf(lg[0], lg[1]), fmaxf(lg[2], lg[3]));
    float ex[FNUM], s = 0.0f;
#pragma unroll
    for (int f = 0; f < FNUM; ++f) { ex[f] = expf(lg[f] - mx); s += ex[f]; }
    float inv = 1.0f / s;
    float sc[FNUM];
#pragma unroll
    for (int f = 0; f < FNUM; ++f) sc[f] = ex[f] * inv;

    float o0 = 0.0f, o1 = 0.0f;
    for (int d = lane; d < HID; d += 32) {
      float hv = h[base + d], v1 = p1[base + d], v2 = p2[base + d];
      float r = 0.0f;
#pragma unroll
      for (int f = 0; f < FNUM; ++f)
        r += sc[f] * (af[f] * hv + be[f] * v1 + ga[f] * v2 + b_filt[f * HID + d]);
      o0 += r * Wc[d * 2 + 0];
      o1 += r * Wc[d * 2 + 1];
    }
    for (int off = 16; off > 0; off >>= 1) {
      o0 += __shfl_xor(o0, off, 32);
      o1 += __shfl_xor(o1, off, 32);
    }
    if (lane == 0) {
      out[(size_t)n * 2 + 0] = o0 + bc[0];
      out[(size_t)n * 2 + 1] = o1 + bc[1];
    }
  }
}

// ---------------------------------------------------------------- launcher
extern "C" void kernel_launch(void* const* d_in, const int* in_sizes, int n_in,
                              void* d_out, int out_size, void* d_ws, size_t ws_size,
                              hipStream_t stream) {
  const float*      x      = (const float*)d_in[0];
  const long long*  ei     = (const long long*)d_in[1];   // int64 [2,E]
  const float*      W1     = (const float*)d_in[2];
  const float*      b1     = (const float*)d_in[3];
  const float*      W2     = (const float*)d_in[4];
  const float*      b2     = (const float*)d_in[5];
  const float*      theta  = (const float*)d_in[6];
  const float*      b_filt = (const float*)d_in[7];
  const float*      Wf     = (const float*)d_in[8];
  const float*      bfv    = (const float*)d_in[9];
  const float*      Wx     = (const float*)d_in[10];
  const float*      bxv    = (const float*)d_in[11];
  const float*      Wc     = (const float*)d_in[12];
  const float*      bc     = (const float*)d_in[13];
  float*            out    = (float*)d_out;

  const int N = in_sizes[0] / IN_F;
  const int E = in_sizes[1] / 2;

  const int PAIRS1 = 84;   // ceil(168/2) for K=166 (last pair zero-padded)
  const int PAIRSH = 78;   // K=156

  float* ws = (float*)d_ws;
  const size_t S = (size_t)N * HID;
  float* dinv = ws;                        // [N]
  float* abc  = ws + N;                    // [12]  (reserve 16)
  float* rowf = ws + N + 16;               // [624] (reserve 640)
  float* Bp1  = ws + N + 656;              // 84*160*2
  float* Bp2  = Bp1 + (size_t)PAIRS1 * NPAD * 2;
  float* Bpf  = Bp2 + (size_t)PAIRSH * NPAD * 2;
  float* Bpx  = Bpf + (size_t)PAIRSH * NPAD * 2;
  float* big  = Bpx + (size_t)PAIRSH * NPAD * 2;
  float* Gh = big + 0 * S;                 // MLP hidden, later h@Wf
  float* h  = big + 1 * S;
  float* p1 = big + 2 * S;
  float* p2 = big + 3 * S;
  float* G1 = big + 4 * S;
  float* G2 = big + 5 * S;
  float* Xp = big + 6 * S;

  // pack weights into WMMA-friendly zero-padded layouts
  pack_b_kernel<<<64, 256, 0, stream>>>(W1, Bp1, IN_F, HID, PAIRS1);
  pack_b_kernel<<<64, 256, 0, stream>>>(W2, Bp2, HID,  HID, PAIRSH);
  pack_b_kernel<<<64, 256, 0, stream>>>(Wf, Bpf, HID,  HID, PAIRSH);
  pack_b_kernel<<<64, 256, 0, stream>>>(Wx, Bpx, HID,  HID, PAIRSH);

  // degrees -> dinv
  fill_zero_kernel<<<256, 256, 0, stream>>>(dinv, (size_t)N);
  degree_kernel<<<1024, 256, 0, stream>>>(ei, dinv, E);
  dinv_kernel<<<(N + 255) / 256, 256, 0, stream>>>(dinv, N);

  // input MLP: h = relu(x@W1+b1)@W2 + b2
  const int gblk = (N + 127) / 128;
  gemm_wmma<<<gblk, 256, 0, stream>>>(x,  IN_F, IN_F, Bp1, b1, Gh, N, 1);
  gemm_wmma<<<gblk, 256, 0, stream>>>(Gh, HID,  HID,  Bp2, b2, h,  N, 0);

  // p1 = A_hat h ; p2 = A_hat p1  (2 props suffice for the K=2 Bernstein basis)
  fill_zero_kernel<<<2048, 256, 0, stream>>>(p1, S);
  prop_kernel<<<2048, 256, 0, stream>>>(h, p1, ei, dinv, E);
  fill_zero_kernel<<<2048, 256, 0, stream>>>(p2, S);
  prop_kernel<<<2048, 256, 0, stream>>>(p1, p2, ei, dinv, E);

  // per-filter scalar coefficients + rowf = b_filt@Wf + bf
  coef_kernel<<<1, 256, 0, stream>>>(theta, b_filt, Wf, bfv, abc, rowf);

  // shared projections (filter GEMMs collapsed by linearity)
  gemm_wmma<<<gblk, 256, 0, stream>>>(h,  HID, HID, Bpf, nullptr, Gh, N, 0);
  gemm_wmma<<<gblk, 256, 0, stream>>>(p1, HID, HID, Bpf, nullptr, G1, N, 0);
  gemm_wmma<<<gblk, 256, 0, stream>>>(p2, HID, HID, Bpf, nullptr, G2, N, 0);
  gemm_wmma<<<gblk, 256, 0, stream>>>(h,  HID, HID, Bpx, bxv,     Xp, N, 0);

  // attention softmax over filters + fusion + classifier head
  final_kernel<<<(N + 7) / 8, 256, 0, stream>>>(Gh, G1, G2, Xp, h, p1, p2,
                                                abc, rowf, b_filt, Wc, bc, out, N);
}